// GatedDeltaRuleModel_65017214926848
// MI455X (gfx1250) — compile-verified
//
#include <hip/hip_runtime.h>

typedef float v2f __attribute__((ext_vector_type(2)));
typedef float v8f __attribute__((ext_vector_type(8)));

#define WMMA_F32(a, b, c) \
  __builtin_amdgcn_wmma_f32_16x16x4_f32(false, (a), false, (b), (short)0, (c), false, false)

constexpr int Bn = 2, Tn = 2048, Hn = 16, DK = 128, DV = 128;
constexpr int CHK = 16;             // chunk length (WMMA tile dim)
constexpr int NCH = Tn / CHK;       // 128 chunks
constexpr int SR  = DK + 4;         // padded LDS row stride (bank-conflict-free column walks)
constexpr int CR  = DK + 4;

// LDS byte offsets (dynamic shared memory starts at LDS offset 0 — no static LDS used)
constexpr unsigned KC_B = (unsigned)(DV * SR) * 4u;          // Kc
constexpr unsigned QC_B = KC_B + (unsigned)(CHK * CR) * 4u;  // Qc
constexpr unsigned VC_B = QC_B + (unsigned)(CHK * CR) * 4u;  // Vc

// Gated delta rule, chunked (UT-transform) form.
// Per chunk (steps s=0..15, cg = inclusive log-decay cumsum, A_s = exp(cg_s)):
//   KS0[t][v] = (S0 k_t)_v                 -> Kc @ S0^T
//   RHS[t]    = beta_t v_t - beta_t A_t KS0[t]
//   M[t][s]   = beta_t exp(cg_t - cg_s) (k_t . k_s),  s < t
//   U         = (I + M)^{-1} RHS
//   O[t]      = A_t (S0 q_t) + sum_{s<=t} exp(cg_t - cg_s)(q_t . k_s) U[s]
//   S0'       = A_15 S0 + sum_s exp(cg_15 - cg_s) U[s] k_s^T
__global__ __launch_bounds__(256, 1) void gdr_chunk_kernel(
    const float* __restrict__ qg, const float* __restrict__ kg,
    const float* __restrict__ vg, const float* __restrict__ gg,
    const float* __restrict__ bg, const float* __restrict__ s0g,
    float* __restrict__ og) {
  extern __shared__ float sm[];
  float* S   = sm;                  // [DV][SR]  persistent state (wave-private row blocks)
  float* Kc  = S   + DV * SR;       // [16][CR]
  float* Qc  = Kc  + CHK * CR;      // [16][CR]
  float* Vc  = Qc  + CHK * CR;      // [16][CR]
  float* KS0 = Vc  + CHK * CR;      // [16][CR]  K@S0^T, overwritten in place by RHS
  float* QS0 = KS0 + CHK * CR;      // [16][CR]  Q@S0^T
  float* Ub  = QS0 + CHK * CR;      // [16][CR]  pseudo-values U
  float* Mm  = Ub  + CHK * CR;      // [16][16]
  float* Ti  = Mm  + 256;           // [16][16]  (I+M)^-1
  float* At  = Ti  + 256;           // [16][16]  causal attn matrix
  float* cg  = At  + 256;           // [16] inclusive cumsum of g
  float* Ai  = cg  + 16;            // [16] exp(cg)
  float* dec = Ai  + 16;            // [16] exp(cg15 - cg)
  float* bet = dec + 16;            // [16]

  const int tid   = threadIdx.x;
  const int lane  = tid & 31, wave = tid >> 5;
  const int lmod  = lane & 15, lhalf = lane >> 4;
  const int bh = blockIdx.x, b = bh / Hn, h = bh % Hn;

  // init state from input (zeros in harness, but load for generality)
  const float* s0 = s0g + (size_t)bh * DV * DK;
  for (int i = tid; i < DV * DK; i += 256)
    S[(i >> 7) * SR + (i & 127)] = s0[i];
  __syncthreads();

  const size_t rowstride = (size_t)Hn * DK;            // elems between consecutive t
  const size_t base0 = ((size_t)b * Tn * Hn + h) * DK; // [b][0][h][0]

  for (int ch = 0; ch < NCH; ++ch) {
    const int t0 = ch * CHK;
    const size_t cb = base0 + (size_t)t0 * rowstride;

    // ---- async DMA chunk of q/k/v straight into LDS (ASYNCcnt path, no VGPR bounce) ----
    // 16 rows x 512B per array; each thread moves two 16B segments per array.
    for (int i = tid; i < 512; i += 256) {
      int t = i >> 5, dq = (i & 31) * 4;          // row, float offset (16B aligned)
      const float* gk = kg + cb + (size_t)t * rowstride + dq;
      const float* gq = qg + cb + (size_t)t * rowstride + dq;
      const float* gv = vg + cb + (size_t)t * rowstride + dq;
      unsigned lk = KC_B + (unsigned)t * (CR * 4u) + (unsigned)dq * 4u;
      asm volatile("global_load_async_to_lds_b128 %0, %1, off"
                   :: "v"(lk), "v"(gk) : "memory");
      asm volatile("global_load_async_to_lds_b128 %0, %1, off"
                   :: "v"(lk + (QC_B - KC_B)), "v"(gq) : "memory");
      asm volatile("global_load_async_to_lds_b128 %0, %1, off"
                   :: "v"(lk + (VC_B - KC_B)), "v"(gv) : "memory");
    }
    if (tid == 0) {  // decay cumsum + gates (16 steps, cheap; overlaps async DMA)
      float acc = 0.f;
      const size_t gb = (size_t)b * Tn * Hn + h;
      for (int i = 0; i < CHK; ++i) {
        size_t gi = gb + (size_t)(t0 + i) * Hn;
        acc += gg[gi];
        cg[i]  = acc;
        Ai[i]  = __expf(acc);
        bet[i] = bg[gi];
      }
      for (int i = 0; i < CHK; ++i) dec[i] = __expf(acc - cg[i]);
    }
    // prefetch next chunk into L2 (global_prefetch_b8): 16 rows x 512B each for q/k/v
    if (ch + 1 < NCH && tid < 128) {
      int t = tid >> 3, seg = tid & 7;
      size_t gi = cb + (size_t)CHK * rowstride + (size_t)t * rowstride + (size_t)seg * 16;
      __builtin_prefetch(kg + gi, 0, 1);
      __builtin_prefetch(qg + gi, 0, 1);
      __builtin_prefetch(vg + gi, 0, 1);
    }
    asm volatile("s_wait_asynccnt 0x0" ::: "memory");
    __syncthreads();

    // ---- KS0 = Kc @ S0^T ; QS0 = Qc @ S0^T ; wave w -> dv columns [16w,16w+16) ----
    {
      const int n0 = wave * 16;
      v8f aK = {}, aQ = {};
      for (int kb = 0; kb < DK; kb += 4) {
        int kk = kb + 2 * lhalf;
        v2f ark; ark.x = Kc[lmod * CR + kk]; ark.y = Kc[lmod * CR + kk + 1];
        v2f arq; arq.x = Qc[lmod * CR + kk]; arq.y = Qc[lmod * CR + kk + 1];
        v2f br;  br.x  = S[(n0 + lmod) * SR + kk]; br.y = S[(n0 + lmod) * SR + kk + 1];
        aK = WMMA_F32(ark, br, aK);
        aQ = WMMA_F32(arq, br, aQ);
      }
      for (int r = 0; r < 8; ++r) {
        int m = r + 8 * lhalf;
        KS0[m * CR + n0 + lmod] = aK[r];
        QS0[m * CR + n0 + lmod] = aQ[r];
      }
    }
    __syncthreads();

    // ---- RHS (in place over KS0), all threads ----
    for (int i = tid; i < CHK * DK; i += 256) {
      int t = i >> 7, d = i & 127;
      KS0[t * CR + d] = bet[t] * (Vc[t * CR + d] - Ai[t] * KS0[t * CR + d]);
    }

    // ---- small 16x16 matrices: wave0 -> M + triangular inverse, wave1 -> Attn ----
    if (wave == 0) {
      v8f acc = {};
      for (int kb = 0; kb < DK; kb += 4) {
        int kk = kb + 2 * lhalf;
        v2f ar; ar.x = Kc[lmod * CR + kk]; ar.y = Kc[lmod * CR + kk + 1];
        acc = WMMA_F32(ar, ar, acc);   // K K^T: A and B operand values coincide
      }
      for (int r = 0; r < 8; ++r) {
        int m = r + 8 * lhalf, n = lmod;
        Mm[m * 16 + n] = (m > n) ? bet[m] * __expf(cg[m] - cg[n]) * acc[r] : 0.f;
      }
      // (I+M)^{-1} by forward substitution; lane n owns column n (no cross-lane deps)
      if (lane < 16) {
        float xcol[16];
        for (int i = 0; i < CHK; ++i) {
          float x = (i == lmod) ? 1.f : 0.f;
          for (int s2 = 0; s2 < i; ++s2) x -= Mm[i * 16 + s2] * xcol[s2];
          xcol[i] = x;
          Ti[i * 16 + lmod] = x;
        }
      }
    } else if (wave == 1) {
      v8f acc = {};
      for (int kb = 0; kb < DK; kb += 4) {
        int kk = kb + 2 * lhalf;
        v2f ar; ar.x = Qc[lmod * CR + kk]; ar.y = Qc[lmod * CR + kk + 1];
        v2f br; br.x = Kc[lmod * CR + kk]; br.y = Kc[lmod * CR + kk + 1];
        acc = WMMA_F32(ar, br, acc);   // Q K^T
      }
      for (int r = 0; r < 8; ++r) {
        int m = r + 8 * lhalf, n = lmod;
        At[m * 16 + n] = (m >= n) ? __expf(cg[m] - cg[n]) * acc[r] : 0.f;
      }
    }
    __syncthreads();

    // ---- U = Ti @ RHS ; wave w -> dv columns [16w,16w+16) ----
    {
      const int n0 = wave * 16;
      v8f acc = {};
      for (int kb = 0; kb < CHK; kb += 4) {
        int kk = kb + 2 * lhalf;
        v2f ar; ar.x = Ti[lmod * 16 + kk];        ar.y = Ti[lmod * 16 + kk + 1];
        v2f br; br.x = KS0[kk * CR + n0 + lmod];  br.y = KS0[(kk + 1) * CR + n0 + lmod];
        acc = WMMA_F32(ar, br, acc);
      }
      for (int r = 0; r < 8; ++r)
        Ub[(r + 8 * lhalf) * CR + n0 + lmod] = acc[r];
    }
    __syncthreads();

    // ---- O = diag(Ai) * QS0 + At @ U ; store chunk outputs ----
    {
      const int n0 = wave * 16;
      v8f acc;
      for (int r = 0; r < 8; ++r) {
        int m = r + 8 * lhalf;
        acc[r] = Ai[m] * QS0[m * CR + n0 + lmod];
      }
      for (int kb = 0; kb < CHK; kb += 4) {
        int kk = kb + 2 * lhalf;
        v2f ar; ar.x = At[lmod * 16 + kk];       ar.y = At[lmod * 16 + kk + 1];
        v2f br; br.x = Ub[kk * CR + n0 + lmod];  br.y = Ub[(kk + 1) * CR + n0 + lmod];
        acc = WMMA_F32(ar, br, acc);
      }
      for (int r = 0; r < 8; ++r) {
        int m = r + 8 * lhalf;
        og[(((size_t)b * Tn + t0 + m) * Hn + h) * DV + n0 + lmod] = acc[r];
      }
    }

    // ---- state update: S = A15*S + U^T diag(dec) Kc (wave-private DV row block) ----
    {
      const int m0 = wave * 16;
      const float Atot = Ai[15];
      for (int j = 0; j < 8; ++j) {
        v8f acc;
        for (int r = 0; r < 8; ++r)
          acc[r] = Atot * S[(m0 + r + 8 * lhalf) * SR + j * 16 + lmod];
        for (int kb = 0; kb < CHK; kb += 4) {
          int kk = kb + 2 * lhalf;
          v2f ar;
          ar.x = dec[kk]     * Ub[kk * CR       + m0 + lmod];
          ar.y = dec[kk + 1] * Ub[(kk + 1) * CR + m0 + lmod];
          v2f br;
          br.x = Kc[kk * CR       + j * 16 + lmod];
          br.y = Kc[(kk + 1) * CR + j * 16 + lmod];
          acc = WMMA_F32(ar, br, acc);
        }
        for (int r = 0; r < 8; ++r)
          S[(m0 + r + 8 * lhalf) * SR + j * 16 + lmod] = acc[r];
      }
    }
    __syncthreads();  // protect Kc/Qc/Vc reload next chunk
  }
}

extern "C" void kernel_launch(void* const* d_in, const int* in_sizes, int n_in,
                              void* d_out, int out_size, void* d_ws, size_t ws_size,
                              hipStream_t stream) {
  const float* q    = (const float*)d_in[0];
  const float* k    = (const float*)d_in[1];
  const float* v    = (const float*)d_in[2];
  const float* g    = (const float*)d_in[3];
  const float* beta = (const float*)d_in[4];
  const float* st   = (const float*)d_in[5];
  float* out = (float*)d_out;

  size_t smem = (size_t)(DV * SR + 6 * CHK * CR + 3 * 256 + 4 * 16) * sizeof(float);
  gdr_chunk_kernel<<<dim3(Bn * Hn), dim3(256), smem, stream>>>(q, k, v, g, beta, st, out);
}